// noflayer_38233798869430
// MI455X (gfx1250) — compile-verified
//
#include <hip/hip_runtime.h>
#include <hip/hip_bf16.h>
#include <math.h>

#define F 128
#define ALPHA 0.2f

typedef float v2f __attribute__((ext_vector_type(2)));
typedef float v8f __attribute__((ext_vector_type(8)));

// ---- monotone ordered-uint encoding of float (for atomic segment-max) ----
__device__ __forceinline__ unsigned ordU(float f) {
    unsigned u = __float_as_uint(f);
    return (u & 0x80000000u) ? ~u : (u | 0x80000000u);
}
__device__ __forceinline__ float ordToF(unsigned v) {
    unsigned u = (v & 0x80000000u) ? (v & 0x7fffffffu) : ~v;
    return __uint_as_float(u);
}

// ---- K0: wvec = [ W_att@a[:F] | W_att@a[F:] | zeros(F) ]  (tiny, one block) ----
__global__ void k_wvec(const float* __restrict__ W, const float* __restrict__ a,
                       float* __restrict__ wvec) {
    int k = threadIdx.x;
    if (k < F) {
        float s1 = 0.f, s2 = 0.f;
        for (int n = 0; n < F; ++n) {
            float w = W[k * F + n];
            s1 += w * a[n];
            s2 += w * a[F + n];
        }
        wvec[k] = s1;          // w1
        wvec[F + k] = s2;      // w2
        wvec[2 * F + k] = 0.f; // zero column source for B cols 2..15
    }
}

// ---- K1: f1 = x@w1, f2 = x@w2 via V_WMMA_F32_16X16X4_F32 ----
// One wave per 16-row tile of x. B = [w1 w2 0 ... 0] (128x16).
// A 16x4 f32 frag: lane l holds row M=l&15, K = k0 + (l>>4)*2 .. +1  (2 VGPRs)
// B 4x16  f32 frag: lane l holds col N=l&15, same K pair            (2 VGPRs)
// C 16x16 f32: VGPR g, lanes 0-15 -> M=g, lanes 16-31 -> M=g+8, N=lane&15
// All loads are unconditional (per-lane source pointer chosen once) so the
// inner loop is 2x global_load_b64 + 1x v_wmma with no EXEC manipulation.
__global__ void k_scores_wmma(const float* __restrict__ x,
                              const float* __restrict__ wvec,
                              float* __restrict__ f1, float* __restrict__ f2, int N) {
    const int wave = (blockIdx.x * blockDim.x + threadIdx.x) >> 5;
    const int ntiles = (N + 15) >> 4;
    if (wave >= ntiles) return;          // whole wave exits together; EXEC stays all-ones
    const int lane = threadIdx.x & 31;
    const int m = lane & 15;             // row-in-tile for A, column for B/C
    const int hi = lane >> 4;
    const int row = wave * 16 + m;
    const int lrow = (row < N) ? row : (N - 1);   // clamp: junk rows are never stored
    const float* xr = x + (size_t)lrow * F + hi * 2;
    const float* bsrc = wvec + ((m == 0) ? 0 : (m == 1) ? F : 2 * F) + hi * 2;

    v8f acc = {0.f, 0.f, 0.f, 0.f, 0.f, 0.f, 0.f, 0.f};
#pragma unroll
    for (int k0 = 0; k0 < F; k0 += 4) {
        v2f afrag = *(const v2f*)(xr + k0);
        v2f bfrag = *(const v2f*)(bsrc + k0);
        acc = __builtin_amdgcn_wmma_f32_16x16x4_f32(
            /*neg_a=*/false, afrag, /*neg_b=*/false, bfrag,
            /*c_mod=*/(short)0, acc, /*reuse_a=*/false, /*reuse_b=*/false);
    }
    // column 0 of C is f1, column 1 is f2
    if (m < 2) {
        float* dst = (m == 0) ? f1 : f2;
#pragma unroll
        for (int g = 0; g < 8; ++g) {
            int r = wave * 16 + hi * 8 + g;
            if (r < N) dst[r] = acc[g];
        }
    }
}

// ---- generic zero-fill ----
__global__ void k_zero(float* __restrict__ p, int n) {
    int i = blockIdx.x * blockDim.x + threadIdx.x;
    if (i < n) p[i] = 0.f;
}

// ---- K2: e = leakyrelu(f1[r]+f2[c]); segment-max via ordered-uint atomicMax ----
__global__ void k_edge_e(const int* __restrict__ rows, const int* __restrict__ cols,
                         const float* __restrict__ f1, const float* __restrict__ f2,
                         float* __restrict__ att, unsigned* __restrict__ mseg, int E) {
    int j = blockIdx.x * blockDim.x + threadIdx.x;
    if (j >= E) return;
    int r = rows[j], c = cols[j];
    float e = f1[r] + f2[c];
    e = (e > 0.f) ? e : ALPHA * e;
    att[j] = e;
    atomicMax(&mseg[r], ordU(e));
}

// ---- K3: ex = exp(e - m[r]); segment-sum s ----
__global__ void k_edge_exp(const int* __restrict__ rows, float* __restrict__ att,
                           const unsigned* __restrict__ mseg, float* __restrict__ s, int E) {
    int j = blockIdx.x * blockDim.x + threadIdx.x;
    if (j >= E) return;
    int r = rows[j];
    float ex = expf(att[j] - ordToF(mseg[r]));
    att[j] = ex;
    atomicAdd(&s[r], ex);
}

// ---- K4: att = ex/s; rowsum += edge_weight * 0.5 * att ----
__global__ void k_edge_norm(const int* __restrict__ rows, const float* __restrict__ ew,
                            float* __restrict__ att, const float* __restrict__ s,
                            float* __restrict__ rsum, int E) {
    int j = blockIdx.x * blockDim.x + threadIdx.x;
    if (j >= E) return;
    int r = rows[j];
    float sv = s[r];
    sv = (sv > 0.f) ? sv : 1.f;
    float at = att[j] / sv;
    att[j] = at;
    atomicAdd(&rsum[r], ew[j] * 0.5f * at);
}

// ---- K5: SpMM hop. One wave per edge, float4 gather, 4 atomic adds per lane.
// update/out are 25.6 MB each -> L2-resident on the 192 MB L2. ----
__global__ void k_spmm(const int* __restrict__ rows, const int* __restrict__ cols,
                       const float* __restrict__ att, const float* __restrict__ in,
                       float* __restrict__ out, int E) {
    int wave = (blockIdx.x * blockDim.x + threadIdx.x) >> 5;
    if (wave >= E) return;
    int lane = threadIdx.x & 31;
    int r = rows[wave], c = cols[wave];
    float at = att[wave];
    const float4* src = (const float4*)(in + (size_t)c * F);
    float4 v = src[lane];
    float* dst = out + (size_t)r * F + lane * 4;
    atomicAdd(dst + 0, at * v.x);
    atomicAdd(dst + 1, at * v.y);
    atomicAdd(dst + 2, at * v.z);
    atomicAdd(dst + 3, at * v.w);
}

// ---- K6: feat_even/feat_odd/feat_prime elementwise ----
__global__ void k_mix(const float* __restrict__ x, const float* __restrict__ U,
                      const float* __restrict__ rsum, const float* __restrict__ temp,
                      float* __restrict__ fp, int NF, int step) {
    int i = blockIdx.x * blockDim.x + threadIdx.x;
    if (i >= NF) return;
    float coe0 = 1.f / (1.f + expf(-temp[0]));
    float coe2 = 1.f / (1.f + expf(-temp[2]));
    int r = i >> 7;                  // i / F
    float u = U[i];
    float fe = coe0 * x[i] + u;
    float fo = u - fe * rsum[r];
    fp[i] = (step == 0) ? fo : (coe2 * fp[i] + (1.f - coe2) * fo);
}

extern "C" void kernel_launch(void* const* d_in, const int* in_sizes, int n_in,
                              void* d_out, int out_size, void* d_ws, size_t ws_size,
                              hipStream_t stream) {
    const float* x    = (const float*)d_in[0];
    // d_in[1] = h0 : unused by the reference
    const int*   ei   = (const int*)d_in[2];
    const float* ew   = (const float*)d_in[3];
    const float* W    = (const float*)d_in[4];
    const float* a    = (const float*)d_in[5];
    const float* temp = (const float*)d_in[6];

    const int N = in_sizes[0] / F;
    const int E = in_sizes[3];
    const int* rows = ei;
    const int* cols = ei + E;
    const int NF = N * F;

    // workspace layout (floats); N and E are multiples of 16 so float4 stays aligned
    float* ws   = (float*)d_ws;
    float* wvec = ws;                      // 3*F : [w1 | w2 | zeros]
    float* f1   = wvec + 3 * F;            // N
    float* f2   = f1 + N;                  // N
    unsigned* mseg = (unsigned*)(f2 + N);  // N  (ordered-uint; 0 == -inf sentinel)
    float* s    = (float*)(mseg + N);      // N
    float* rsum = s + N;                   // N
    float* att  = rsum + N;                // E
    float* Ua   = att + E;                 // N*F
    float* Ub   = Ua + (size_t)NF;         // N*F
    float* out  = (float*)d_out;

    const int T = 256;
    const int ntiles = (N + 15) / 16;

    // attention scores
    k_wvec<<<1, 128, 0, stream>>>(W, a, wvec);
    k_scores_wmma<<<(ntiles + 7) / 8, T, 0, stream>>>(x, wvec, f1, f2, N);

    // zero m/s/rowsum (contiguous 3N region) and run edge passes
    k_zero<<<(3 * N + T - 1) / T, T, 0, stream>>>((float*)mseg, 3 * N);
    k_edge_e   <<<(E + T - 1) / T, T, 0, stream>>>(rows, cols, f1, f2, att, mseg, E);
    k_edge_exp <<<(E + T - 1) / T, T, 0, stream>>>(rows, att, mseg, s, E);
    k_edge_norm<<<(E + T - 1) / T, T, 0, stream>>>(rows, ew, att, s, rsum, E);

    // 3 hops: zero accumulator, SpMM, elementwise mix into d_out
    const int spmmBlocks = (E * 32 + T - 1) / T;   // one wave (32 lanes) per edge
    const int nfBlocks   = (NF + T - 1) / T;

    k_zero<<<nfBlocks, T, 0, stream>>>(Ua, NF);
    k_spmm<<<spmmBlocks, T, 0, stream>>>(rows, cols, att, x, Ua, E);
    k_mix <<<nfBlocks, T, 0, stream>>>(x, Ua, rsum, temp, out, NF, 0);

    k_zero<<<nfBlocks, T, 0, stream>>>(Ub, NF);
    k_spmm<<<spmmBlocks, T, 0, stream>>>(rows, cols, att, Ua, Ub, E);
    k_mix <<<nfBlocks, T, 0, stream>>>(x, Ub, rsum, temp, out, NF, 1);

    k_zero<<<nfBlocks, T, 0, stream>>>(Ua, NF);
    k_spmm<<<spmmBlocks, T, 0, stream>>>(rows, cols, att, Ub, Ua, E);
    k_mix <<<nfBlocks, T, 0, stream>>>(x, Ua, rsum, temp, out, NF, 2);
}